// CoordAtt_82119774699955
// MI455X (gfx1250) — compile-verified
//
#include <hip/hip_runtime.h>
#include <hip/hip_bf16.h>
#include <math.h>

typedef __attribute__((ext_vector_type(2))) float v2f;
typedef __attribute__((ext_vector_type(8))) float v8f;

#define BN_EPS 1e-5f

// ---------------------------------------------------------------------------
// Kernel 1: H/W mean pooling.  y0[bc, p] : p<96 -> mean over w of row p,
//                              p>=96     -> mean over h of column p-96.
// One block per (b,c); 96x96 fp32 tile staged in LDS (36 KB of 320 KB/WGP).
// ---------------------------------------------------------------------------
__global__ void pool_kernel(const float* __restrict__ x, float* __restrict__ y0) {
  __shared__ float tile[96 * 96];
  const int bc = blockIdx.x;
  const float* xp = x + (size_t)bc * 9216;
  for (int i = threadIdx.x; i < 9216; i += blockDim.x) tile[i] = xp[i];
  __syncthreads();
  const int t = threadIdx.x;
  if (t < 96) {
    float rs = 0.f, cs = 0.f;
#pragma unroll 8
    for (int j = 0; j < 96; ++j) {
      rs += tile[t * 96 + j];
      cs += tile[j * 96 + t];
    }
    y0[(size_t)bc * 192 + t]      = rs * (1.0f / 96.0f);
    y0[(size_t)bc * 192 + 96 + t] = cs * (1.0f / 96.0f);
  }
}

// ---------------------------------------------------------------------------
// Generic batched conv1x1 as WMMA f32 GEMM:
//   Out[b, o, n] = act( sum_c W[o,c] * In[b, c, n] + bias[o] )
// One wave32 per 16x16 output tile; K stepped by 4 via v_wmma_f32_16x16x4_f32.
// A layout: lane l holds M = m0 + (l&15); VGPR v holds K = k0 + 2*(l>>4) + v.
// B layout (mirror):  lane l holds N = n0 + (l&15); VGPR v holds same K.
// C/D layout: VGPR v holds M = m0 + v + 8*(l>>4), N = n0 + (l&15).
// act: 0 = none, 1 = sigmoid.
// ---------------------------------------------------------------------------
__global__ void gemm16_wmma(const float* __restrict__ W,
                            const float* __restrict__ bias,
                            const float* __restrict__ In,
                            float* __restrict__ Out,
                            int K, int inBatch, int inRow,
                            int outBatch, int outRow, int act) {
  const int n0 = blockIdx.x * 16;
  const int m0 = blockIdx.y * 16;
  const int b  = blockIdx.z;
  const int lane = threadIdx.x;
  const int lm = lane & 15;
  const int lh = lane >> 4;

  const float* wrow = W + (size_t)(m0 + lm) * K + 2 * lh;          // A source
  const float* bcol = In + (size_t)b * inBatch + n0 + lm;          // B source

  v8f acc = {};
  for (int k0 = 0; k0 < K; k0 += 4) {
    v2f a, bb;
    a[0] = wrow[k0];
    a[1] = wrow[k0 + 1];
    const int kb = k0 + 2 * lh;
    bb[0] = bcol[(size_t)kb * inRow];
    bb[1] = bcol[(size_t)(kb + 1) * inRow];
    acc = __builtin_amdgcn_wmma_f32_16x16x4_f32(
        /*neg_a=*/false, a, /*neg_b=*/false, bb,
        /*c_mod=*/(short)0, acc, /*reuse_a=*/false, /*reuse_b=*/false);
  }

#pragma unroll
  for (int v = 0; v < 8; ++v) {
    const int row = m0 + v + 8 * lh;
    float val = acc[v] + bias[row];
    if (act == 1) val = 1.0f / (1.0f + __expf(-val));
    Out[(size_t)b * outBatch + (size_t)row * outRow + n0 + lm] = val;
  }
}

// ---------------------------------------------------------------------------
// Dynamic-weight channels actually needed by the involution.
// Singleton spatial dim => only taps k = 7*kh+3 (mode 0) or k = 21+kw (mode 1)
// survive the zero padding.  wg[b, g, j, n] = b2[row] + sum_r w2[row,r]*t[b,r,n]
// with row = g*49 + k(j).   grid = (28, B)
// ---------------------------------------------------------------------------
__global__ void wg_kernel(const float* __restrict__ w2, const float* __restrict__ b2,
                          const float* __restrict__ t, float* __restrict__ wg,
                          int N, int mode) {
  const int gj = blockIdx.x;           // g*7 + j
  const int b  = blockIdx.y;
  const int g = gj / 7, j = gj % 7;
  const int k = (mode == 0) ? (j * 7 + 3) : (21 + j);
  const int row = g * 49 + k;
  const float bval = b2[row];
  for (int n = threadIdx.x; n < N; n += blockDim.x) {
    float acc = bval;
#pragma unroll
    for (int r = 0; r < 16; ++r)
      acc += w2[row * 16 + r] * t[((size_t)b * 16 + r) * N + n];
    wg[((size_t)b * 28 + gj) * N + n] = acc;
  }
}

// ---------------------------------------------------------------------------
// 7-tap 1-D dynamic conv along n (zero padded), optional fused BN + hswish.
//   out[b, c, n] = sum_j wg[b, c/16, j, n] * in[b, c, n + j - 3]
// grid = (64, B).  For the first involution this runs over the full
// concatenated 192-axis, so the h/w halves correctly bleed at the seam.
// ---------------------------------------------------------------------------
__global__ void combine_kernel(const float* __restrict__ in, int inBatch, int inRow,
                               const float* __restrict__ wg, float* __restrict__ out,
                               int N,
                               const float* __restrict__ bng, const float* __restrict__ bnb,
                               const float* __restrict__ bnm, const float* __restrict__ bnv,
                               int do_bn) {
  const int c = blockIdx.x;
  const int b = blockIdx.y;
  const int g = c >> 4;
  const float* inp = in + (size_t)b * inBatch + (size_t)c * inRow;
  const float* wgp = wg + ((size_t)b * 28 + g * 7) * N;

  float scale = 1.f, shift = 0.f;
  if (do_bn) {
    scale = bng[c] * rsqrtf(bnv[c] + BN_EPS);
    shift = bnb[c] - bnm[c] * scale;
  }
  for (int n = threadIdx.x; n < N; n += blockDim.x) {
    float acc = 0.f;
#pragma unroll
    for (int j = 0; j < 7; ++j) {
      const int m = n + j - 3;
      const float v = (m >= 0 && m < N) ? inp[m] : 0.0f;
      acc += wgp[j * N + n] * v;
    }
    if (do_bn) {
      acc = acc * scale + shift;
      float t6 = fminf(fmaxf(acc + 3.0f, 0.0f), 6.0f);
      acc = acc * t6 * (1.0f / 6.0f);
    }
    out[((size_t)b * 64 + c) * N + n] = acc;
  }
}

// ---------------------------------------------------------------------------
// Final: out = x * a_w * a_h, vectorized float4 (24 float4 per 96-wide row).
// ---------------------------------------------------------------------------
__global__ void final_mul(const float4* __restrict__ x,
                          const float* __restrict__ a_h,
                          const float* __restrict__ a_w,
                          float4* __restrict__ out, int total4) {
  const int idx = blockIdx.x * blockDim.x + threadIdx.x;
  if (idx >= total4) return;
  const int w4 = idx % 24;
  const int rest = idx / 24;
  const int h = rest % 96;
  const int bc = rest / 96;
  const float ah = a_h[(size_t)bc * 96 + h];
  const float4 aw = *(const float4*)&a_w[(size_t)bc * 96 + w4 * 4];
  float4 v = x[idx];
  float4 o;
  o.x = v.x * aw.x * ah;
  o.y = v.y * aw.y * ah;
  o.z = v.z * aw.z * ah;
  o.w = v.w * aw.w * ah;
  out[idx] = o;
}

// ---------------------------------------------------------------------------
extern "C" void kernel_launch(void* const* d_in, const int* in_sizes, int n_in,
                              void* d_out, int out_size, void* d_ws, size_t ws_size,
                              hipStream_t stream) {
  (void)in_sizes; (void)n_in; (void)out_size; (void)ws_size;

  const float* x       = (const float*)d_in[0];
  const float* w_c1    = (const float*)d_in[1];
  const float* b_c1    = (const float*)d_in[2];
  const float* inv1_w1 = (const float*)d_in[3];
  const float* inv1_b1 = (const float*)d_in[4];
  const float* inv1_w2 = (const float*)d_in[5];
  const float* inv1_b2 = (const float*)d_in[6];
  const float* bn_g    = (const float*)d_in[7];
  const float* bn_b    = (const float*)d_in[8];
  const float* bn_m    = (const float*)d_in[9];
  const float* bn_v    = (const float*)d_in[10];
  const float* invh_w1 = (const float*)d_in[11];
  const float* invh_b1 = (const float*)d_in[12];
  const float* invh_w2 = (const float*)d_in[13];
  const float* invh_b2 = (const float*)d_in[14];
  const float* w_h     = (const float*)d_in[15];
  const float* b_h     = (const float*)d_in[16];
  const float* invw_w1 = (const float*)d_in[17];
  const float* invw_b1 = (const float*)d_in[18];
  const float* invw_w2 = (const float*)d_in[19];
  const float* invw_b2 = (const float*)d_in[20];
  const float* w_w     = (const float*)d_in[21];
  const float* b_w     = (const float*)d_in[22];
  float* out = (float*)d_out;

  // workspace layout (floats)
  float* ws = (float*)d_ws;
  size_t o = 0;
  float* y0  = ws + o; o += 16 * 256 * 192;   // pooled concat
  float* y1  = ws + o; o += 16 * 64 * 192;    // conv1 out
  float* t1  = ws + o; o += 16 * 16 * 192;    // inv1 reduce
  float* wg1 = ws + o; o += 16 * 28 * 192;    // inv1 dyn weights (4g x 7taps)
  float* yb  = ws + o; o += 16 * 64 * 192;    // inv1 + BN + hswish
  float* th  = ws + o; o += 16 * 16 * 96;
  float* wgh = ws + o; o += 16 * 28 * 96;
  float* zh  = ws + o; o += 16 * 64 * 96;
  float* tw  = ws + o; o += 16 * 16 * 96;
  float* wgw = ws + o; o += 16 * 28 * 96;
  float* zw  = ws + o; o += 16 * 64 * 96;
  float* ah  = ws + o; o += 16 * 256 * 96;    // sigmoid attention (h)
  float* aw  = ws + o; o += 16 * 256 * 96;    // sigmoid attention (w)

  // 1) pooling: 4096 (b,c) tiles
  pool_kernel<<<dim3(16 * 256), dim3(128), 0, stream>>>(x, y0);

  // 2) conv1: M=64, K=256, N=192, B=16
  gemm16_wmma<<<dim3(12, 4, 16), dim3(32), 0, stream>>>(
      w_c1, b_c1, y0, y1, 256, 256 * 192, 192, 64 * 192, 192, 0);

  // 3) involution 1 on concatenated axis (N=192)
  gemm16_wmma<<<dim3(12, 1, 16), dim3(32), 0, stream>>>(
      inv1_w1, inv1_b1, y1, t1, 64, 64 * 192, 192, 16 * 192, 192, 0);
  wg_kernel<<<dim3(28, 16), dim3(96), 0, stream>>>(inv1_w2, inv1_b2, t1, wg1, 192, 0);
  combine_kernel<<<dim3(64, 16), dim3(192), 0, stream>>>(
      y1, 64 * 192, 192, wg1, yb, 192, bn_g, bn_b, bn_m, bn_v, 1);

  // 4) h-branch: y_h = yb[:, :, 0:96]
  gemm16_wmma<<<dim3(6, 1, 16), dim3(32), 0, stream>>>(
      invh_w1, invh_b1, yb, th, 64, 64 * 192, 192, 16 * 96, 96, 0);
  wg_kernel<<<dim3(28, 16), dim3(96), 0, stream>>>(invh_w2, invh_b2, th, wgh, 96, 0);
  combine_kernel<<<dim3(64, 16), dim3(96), 0, stream>>>(
      yb, 64 * 192, 192, wgh, zh, 96, nullptr, nullptr, nullptr, nullptr, 0);
  gemm16_wmma<<<dim3(6, 16, 16), dim3(32), 0, stream>>>(
      w_h, b_h, zh, ah, 64, 64 * 96, 96, 256 * 96, 96, 1);

  // 5) w-branch: y_w = yb[:, :, 96:192] (transpose is a no-op on flat layout)
  gemm16_wmma<<<dim3(6, 1, 16), dim3(32), 0, stream>>>(
      invw_w1, invw_b1, yb + 96, tw, 64, 64 * 192, 192, 16 * 96, 96, 0);
  wg_kernel<<<dim3(28, 16), dim3(96), 0, stream>>>(invw_w2, invw_b2, tw, wgw, 96, 1);
  combine_kernel<<<dim3(64, 16), dim3(96), 0, stream>>>(
      yb + 96, 64 * 192, 192, wgw, zw, 96, nullptr, nullptr, nullptr, nullptr, 0);
  gemm16_wmma<<<dim3(6, 16, 16), dim3(32), 0, stream>>>(
      w_w, b_w, zw, aw, 64, 64 * 96, 96, 256 * 96, 96, 1);

  // 6) out = x * a_w * a_h  (float4: 9437184 vectors)
  const int total4 = 16 * 256 * 96 * 24;
  final_mul<<<dim3((total4 + 255) / 256), dim3(256), 0, stream>>>(
      (const float4*)x, ah, aw, (float4*)out, total4);
}